// GCN_12962211299622
// MI455X (gfx1250) — compile-verified
//
#include <hip/hip_runtime.h>
#include <hip/hip_bf16.h>

typedef float v2f __attribute__((ext_vector_type(2)));
typedef float v8f __attribute__((ext_vector_type(8)));

#define HF 256          // feature width (IN_FEATS == H_FEATS == 256)
#define NC 64           // num classes
#define LDS_STRIDE 260  // 256 + 4 pad -> conflict-free A-fragment banks

// ---------------- degree counts (int) ----------------
__global__ void gcn_degrees(const int* __restrict__ src, const int* __restrict__ dst,
                            int* __restrict__ cntO, int* __restrict__ cntI, int E) {
    int e = blockIdx.x * blockDim.x + threadIdx.x;
    if (e >= E) return;
    atomicAdd(&cntO[src[e]], 1);
    atomicAdd(&cntI[dst[e]], 1);
}

// ---------------- single-block exclusive scan over cntI -> offsets, cursor ----------------
__global__ void gcn_scan(const int* __restrict__ cnt, int* __restrict__ offsets,
                         int* __restrict__ cursor, int N) {
    __shared__ int sums[1024];
    int t = threadIdx.x;
    int chunk = (N + 1023) / 1024;
    int start = t * chunk;
    int end   = start + chunk; if (end > N) end = N; if (start > N) start = N;
    int s = 0;
    for (int i = start; i < end; ++i) s += cnt[i];
    sums[t] = s;
    __syncthreads();
    for (int off = 1; off < 1024; off <<= 1) {   // inclusive Hillis-Steele
        int v = (t >= off) ? sums[t - off] : 0;
        __syncthreads();
        if (t >= off) sums[t] += v;
        __syncthreads();
    }
    int run = (t == 0) ? 0 : sums[t - 1];
    for (int i = start; i < end; ++i) {
        offsets[i] = run;
        cursor[i]  = run;
        run += cnt[i];
    }
}

// ---------------- CSR fill: bucket src ids by dst ----------------
__global__ void gcn_fill(const int* __restrict__ src, const int* __restrict__ dst,
                         int* __restrict__ cursor, int* __restrict__ srcSorted, int E) {
    int e = blockIdx.x * blockDim.x + threadIdx.x;
    if (e >= E) return;
    int pos = atomicAdd(&cursor[dst[e]], 1);
    srcSorted[pos] = src[e];
}

// ---------------- W[k][n] -> Wt[n][k] (makes B-fragments contiguous b64 loads) ----------------
__global__ void gcn_transpose(const float* __restrict__ W, float* __restrict__ Wt) {
    int n = blockIdx.x;     // 256 blocks
    int k = threadIdx.x;    // 256 threads
    Wt[(size_t)n * HF + k] = W[(size_t)k * HF + n];
}

// ---------------- h = (x * rsqrt(max(out_deg,1))) @ W  via f32 WMMA ----------------
// 512 threads = 16 waves; block owns a 16-row strip, wave w owns cols [16w,16w+16).
__global__ void gcn_gemm1(const float* __restrict__ x, const float* __restrict__ Wt,
                          const int* __restrict__ cntO, float* __restrict__ h, int N) {
    __shared__ float lds[16 * LDS_STRIDE];
    const int row0 = blockIdx.x * 16;
    const int t = threadIdx.x;

    for (int i = t; i < 16 * (HF / 4); i += blockDim.x) {
        int r  = i >> 6;
        int c4 = i & 63;
        int rr = row0 + r; if (rr >= N) rr = N - 1;
        float4 v = ((const float4*)(x + (size_t)rr * HF))[c4];
        int dg = cntO[rr];
        float rs = rsqrtf((float)(dg > 1 ? dg : 1));
        v.x *= rs; v.y *= rs; v.z *= rs; v.w *= rs;
        *(float4*)(&lds[r * LDS_STRIDE + c4 * 4]) = v;
    }
    __syncthreads();

    const int wave = t >> 5;
    const int lane = t & 31;
    const int m    = lane & 15;
    const int half = lane >> 4;     // selects K pair {0,1} vs {2,3}
    const int col0 = wave * 16;

    const float* Arow = &lds[m * LDS_STRIDE + 2 * half];
    const v2f*   Bt   = (const v2f*)(Wt + (size_t)(col0 + m) * HF);  // Wt[n][k], kb even

    v8f acc = {};
#pragma unroll 8
    for (int k = 0; k < HF; k += 4) {
        v2f a; a.x = Arow[k]; a.y = Arow[k + 1];
        v2f b = Bt[(k + 2 * half) >> 1];
        acc = __builtin_amdgcn_wmma_f32_16x16x4_f32(false, a, false, b,
                                                    (short)0, acc, false, false);
    }

    // C/D layout: VGPR r, lane l -> M = r + 8*(l/16), N = l%16
    float* hp = h + (size_t)(row0 + 8 * half) * HF + col0 + m;
    if (row0 + 16 <= N) {           // uniform fast path (N % 16 == 0 here)
#pragma unroll
        for (int r = 0; r < 8; ++r) hp[(size_t)r * HF] = acc[r];
    } else {
#pragma unroll
        for (int r = 0; r < 8; ++r)
            if (row0 + r + 8 * half < N) hp[(size_t)r * HF] = acc[r];
    }
}

// ---------------- fused gather-aggregate + norm + bias + relu + mean-pool ----------------
// 256 threads = 8 waves; one wave per dst node; lane owns 8 consecutive columns.
__global__ void gcn_aggregate(const int* __restrict__ offsets, const int* __restrict__ srcSorted,
                              const int* __restrict__ cntI, const float* __restrict__ h,
                              const float* __restrict__ bvec, float* __restrict__ pooled, int N) {
    __shared__ float part[8][HF];
    const int t    = threadIdx.x;
    const int w    = t >> 5;
    const int lane = t & 31;
    const int node = blockIdx.x * 8 + w;

    float acc[8] = {0.f, 0.f, 0.f, 0.f, 0.f, 0.f, 0.f, 0.f};
    if (node < N) {
        const int j0 = offsets[node];
        const int j1 = j0 + cntI[node];
        for (int j = j0; j < j1; ++j) {
            int s = srcSorted[j];                       // uniform across wave
            const float4* hp = (const float4*)(h + (size_t)s * HF) + lane * 2;
            float4 v0 = hp[0];
            float4 v1 = hp[1];
            acc[0] += v0.x; acc[1] += v0.y; acc[2] += v0.z; acc[3] += v0.w;
            acc[4] += v1.x; acc[5] += v1.y; acc[6] += v1.z; acc[7] += v1.w;
        }
        int dg = cntI[node];
        float rs = rsqrtf((float)(dg > 1 ? dg : 1));
#pragma unroll
        for (int q = 0; q < 8; ++q) {
            int c = lane * 8 + q;
            float v = acc[q] * rs + bvec[c];
            part[w][c] = (v > 0.f) ? v : 0.f;
        }
    } else {
#pragma unroll
        for (int q = 0; q < 8; ++q) part[w][lane * 8 + q] = 0.f;
    }
    __syncthreads();

    float sum = 0.f;
#pragma unroll
    for (int w2 = 0; w2 < 8; ++w2) sum += part[w2][t];
    atomicAdd(&pooled[t], sum);
}

// ---------------- logits = (pooled/N) @ W2 + b2; softmax(64) ----------------
__global__ void gcn_head(const float* __restrict__ pooled, const float* __restrict__ W2,
                         const float* __restrict__ b2, float* __restrict__ out, int N) {
    __shared__ float logits[NC];
    __shared__ float ex[NC];
    int j = threadIdx.x;
    float invN = 1.0f / (float)N;
    float acc = b2[j];
    for (int k = 0; k < HF; ++k)
        acc += pooled[k] * invN * W2[k * NC + j];
    logits[j] = acc;
    __syncthreads();
    float mx = -3.402823466e38f;
    for (int k = 0; k < NC; ++k) mx = fmaxf(mx, logits[k]);
    float e = __expf(acc - mx);
    ex[j] = e;
    __syncthreads();
    float s = 0.0f;
    for (int k = 0; k < NC; ++k) s += ex[k];
    out[j] = e / s;
}

extern "C" void kernel_launch(void* const* d_in, const int* in_sizes, int n_in,
                              void* d_out, int out_size, void* d_ws, size_t ws_size,
                              hipStream_t stream) {
    const float* x   = (const float*)d_in[0];
    const int*   src = (const int*)  d_in[1];
    const int*   dst = (const int*)  d_in[2];
    const float* W   = (const float*)d_in[3];
    const float* b   = (const float*)d_in[4];
    const float* W2  = (const float*)d_in[5];
    const float* b2  = (const float*)d_in[6];
    float* out = (float*)d_out;

    const int N = in_sizes[0] / HF;   // 100000
    const int E = in_sizes[1];        // 1600000

    // workspace layout (dword units)
    int*   cntO      = (int*)d_ws;                       // [N]
    int*   cntI      = cntO + N;                         // [N]
    int*   offsets   = cntI + N;                         // [N]
    int*   cursor    = offsets + N;                      // [N]
    int*   srcSorted = cursor + N;                       // [E]
    float* pooled    = (float*)(srcSorted + E);          // [256]
    float* Wt        = pooled + HF;                      // [256*256]
    float* h         = Wt + (size_t)HF * HF;             // [N,256]

    hipMemsetAsync(cntO, 0, (size_t)2 * N * sizeof(int), stream);   // cntO + cntI
    hipMemsetAsync(pooled, 0, HF * sizeof(float), stream);

    gcn_degrees  <<<(E + 255) / 256, 256, 0, stream>>>(src, dst, cntO, cntI, E);
    gcn_scan     <<<1, 1024, 0, stream>>>(cntI, offsets, cursor, N);
    gcn_fill     <<<(E + 255) / 256, 256, 0, stream>>>(src, dst, cursor, srcSorted, E);
    gcn_transpose<<<HF, HF, 0, stream>>>(W, Wt);
    gcn_gemm1    <<<(N + 15) / 16, 512, 0, stream>>>(x, Wt, cntO, h, N);
    gcn_aggregate<<<(N + 7) / 8, 256, 0, stream>>>(offsets, srcSorted, cntI, h, b, pooled, N);
    gcn_head     <<<1, NC, 0, stream>>>(pooled, W2, b2, out, N);
}